// SemlaLayer_11931419148532
// MI455X (gfx1250) — compile-verified
//
#include <hip/hip_runtime.h>

// ---------------------------------------------------------------------------
// SemlaLayer fused attention for gfx1250 (MI455X).
//  * Bandwidth-bound: messages (~168 MB) are read exactly once, staged into
//    LDS with GLOBAL_LOAD_ASYNC_TO_LDS_B128 (ASYNCcnt / s_wait_asynccnt).
//  * Softmax done as 2-pass over the LDS tile (1 exp per element, no serial
//    rescale chain); equi uses 2 chunks merged with one closed-form rescale.
//  * Channel-mixing GEMMs use V_WMMA_F32_16X16X4_F32 (fp32 matrix cores).
// ---------------------------------------------------------------------------

#define BB 8
#define NN 256
#define CE 64     // D_EQUI
#define DI 256    // D_INV
#define NH 16
#define DH 16
#define NEG_BIG (-1.0e30f)

typedef __attribute__((ext_vector_type(2))) float v2f;
typedef __attribute__((ext_vector_type(8))) float v8f;
typedef __attribute__((ext_vector_type(4))) int   v4i;

#if __has_builtin(__builtin_amdgcn_global_load_async_to_lds_b128) && \
    __has_builtin(__builtin_amdgcn_s_wait_asynccnt)
#define HAVE_ASYNC 1
#else
#define HAVE_ASYNC 0
#endif

#define ASG __attribute__((address_space(1)))
#define ASL __attribute__((address_space(3)))

// ---------------------------------------------------------------------------
// Generic fp32 WMMA GEMM:  out[M,Nout] = A[M,K] * W[Nout,K]^T (+ bias)
// One 16x16 output tile per wave32; K stepped by 4 (V_WMMA_F32_16X16X4_F32).
// Grid must supply exactly (M/16)*(Nout/16) waves.
//
// fp32 WMMA VGPR layouts (CDNA5 ISA 7.12.2):
//  A 16x4 : lane&15 = M; VGPR0 holds K = k0 + (lane<16 ? 0:2), VGPR1 = +1
//  B 4x16 : lane&15 = N; same K striping as A
//  C/D    : VGPR j -> (M = j + (lane<16 ? 0:8), N = lane&15)
// ---------------------------------------------------------------------------
__global__ void gemm_wmma_f32(const float* __restrict__ A,
                              const float* __restrict__ W,
                              const float* __restrict__ bias,
                              float* __restrict__ out,
                              int M, int K, int Nout) {
    const int wave   = (blockIdx.x * blockDim.x + threadIdx.x) >> 5;
    const int lane   = threadIdx.x & 31;
    const int tilesN = Nout >> 4;
    const int tM     = wave / tilesN;
    const int tN     = wave - tM * tilesN;

    const int m15  = lane & 15;
    const int koff = (lane < 16) ? 0 : 2;
    const int row  = tM * 16 + m15;   // A row this lane feeds
    const int col  = tN * 16 + m15;   // B column (== W row) this lane feeds

    const float* __restrict__ Ap = A + (size_t)row * K + koff;
    const float* __restrict__ Wp = W + (size_t)col * K + koff;

    v8f c = {};
    for (int k0 = 0; k0 < K; k0 += 4) {
        v2f a, b;
        a[0] = Ap[k0];
        a[1] = Ap[k0 + 1];
        b[0] = Wp[k0];
        b[1] = Wp[k0 + 1];
        c = __builtin_amdgcn_wmma_f32_16x16x4_f32(
            /*neg_a=*/false, a, /*neg_b=*/false, b,
            /*c_mod=*/(short)0, c, /*reuse_a=*/false, /*reuse_b=*/false);
    }

    const int outRow = tM * 16 + ((lane < 16) ? 0 : 8);
    const int outCol = tN * 16 + m15;
    const float bv = bias ? bias[outCol] : 0.0f;
#pragma unroll
    for (int j = 0; j < 8; ++j)
        out[(size_t)(outRow + j) * Nout + outCol] = c[j] + bv;
}

// ---------------------------------------------------------------------------
// Equi attention core. One block per (b,q), 64 threads (thread = channel c).
// Messages staged to LDS in two 128-row chunks (32 KB each) via async loads;
// per-chunk 2-pass softmax, chunks merged with one closed-form rescale.
// out_pre[b,q,d,c] = acc_d * sqrt(sq) / s^2    (attn unnormalized exp sums)
// ---------------------------------------------------------------------------
__global__ __launch_bounds__(CE) void
equi_core(const float* __restrict__ emsg,   // [B,N,N,CE]
          const float* __restrict__ projE,  // [B,N,3,CE]
          const int*   __restrict__ adj,    // [B,N,N]
          float* __restrict__ out_pre) {    // [B,N,3,CE]
    const int bq = blockIdx.x;            // b*256 + q
    const int c  = threadIdx.x;           // 0..63

    __shared__ float bias[NN];
    __shared__ float tile[128 * CE];      // 32 KB chunk
    __shared__ int   anyFlag;
    if (c == 0) anyFlag = 0;
    __syncthreads();
    int la = 0;
    for (int i = c; i < NN; i += CE) {
        const int a = adj[(size_t)bq * NN + i];
        bias[i] = a ? 0.0f : NEG_BIG;
        la |= a;
    }
    if (la) atomicOr(&anyFlag, 1);
    __syncthreads();
    if (anyFlag == 0) {                    // fully disconnected row -> mask 0
        for (int i = c; i < NN; i += CE) bias[i] = 0.0f;
    }
    __syncthreads();

    const float* __restrict__ pp = projE + (size_t)(bq >> 8) * NN * 3 * CE + c;
    const char*  gbase = (const char*)(emsg + (size_t)bq * NN * CE);

    float M2[2], S2[2], SQ2[2], A0[2], A1[2], A2[2];

#pragma unroll
    for (int half = 0; half < 2; ++half) {
        const int k0 = half * 128;
        // ---- stage 128 rows (32 KB) into LDS ----
#if HAVE_ASYNC
        for (int i = 0; i < 32; ++i) {
            const int p = i * CE + c;     // 16-byte piece index within chunk
            __builtin_amdgcn_global_load_async_to_lds_b128(
                (ASG v4i*)(gbase + (size_t)k0 * CE * 4 + (size_t)p * 16),
                (ASL v4i*)(tile + p * 4), 0, 0);
        }
        __builtin_amdgcn_s_wait_asynccnt(0);
#else
        for (int i = 0; i < 32; ++i) {
            const int p = i * CE + c;
            ((float4*)tile)[p] =
                ((const float4*)(gbase + (size_t)k0 * CE * 4))[p];
        }
#endif
        __syncthreads();

        // ---- pass 1: per-channel max (LDS bank-conflict free) ----
        float mx = NEG_BIG;
        for (int kk = 0; kk < 128; ++kk)
            mx = fmaxf(mx, tile[kk * CE + c] + bias[k0 + kk]);

        // ---- pass 2: exp-sum + weighted accumulation over proj ----
        float s = 0.0f, sq = 0.0f, a0 = 0.0f, a1 = 0.0f, a2 = 0.0f;
        const float* __restrict__ pk = pp + (size_t)k0 * 3 * CE;
        for (int kk = 0; kk < 128; ++kk) {
            const float em = __expf(tile[kk * CE + c] + bias[k0 + kk] - mx);
            const float* __restrict__ q = pk + (size_t)kk * (3 * CE);
            __builtin_prefetch(q + 8 * 3 * CE, 0, 0);
            s  += em;
            sq += em * em;
            a0 += em * q[0];
            a1 += em * q[CE];
            a2 += em * q[2 * CE];
        }
        M2[half] = mx; S2[half] = s; SQ2[half] = sq;
        A0[half] = a0; A1[half] = a1; A2[half] = a2;
        __syncthreads();                  // all reads done before next chunk
    }

    // ---- merge the two chunk softmaxes ----
    const float mg = fmaxf(M2[0], M2[1]);
    const float c0 = __expf(M2[0] - mg);
    const float c1 = __expf(M2[1] - mg);
    const float s  = S2[0] * c0 + S2[1] * c1;
    const float sq = SQ2[0] * c0 * c0 + SQ2[1] * c1 * c1;
    const float f  = sqrtf(sq) / (s * s);

    float* __restrict__ o = out_pre + (size_t)bq * 3 * CE + c;
    o[0]      = (A0[0] * c0 + A0[1] * c1) * f;
    o[CE]     = (A1[0] * c0 + A1[1] * c1) * f;
    o[2 * CE] = (A2[0] * c0 + A2[1] * c1) * f;
}

// ---------------------------------------------------------------------------
// Inv attention core. One block per (b,q), 256 threads.
// 16 KB message tile staged via async loads. Softmax stats computed with
// strip-parallel reductions (exp evaluated exactly once per element, written
// back into the tile), then phase-B weighted accumulation per (h,e) thread.
// ---------------------------------------------------------------------------
__global__ __launch_bounds__(NN) void
inv_core(const float* __restrict__ imsg,   // [B,N,N,NH]
         const float* __restrict__ projI,  // [B,N,DI]
         const int*   __restrict__ adj,    // [B,N,N]
         float* __restrict__ out_pre) {    // [B,N,DI]
    const int bq = blockIdx.x;
    const int t  = threadIdx.x;            // 0..255

    __shared__ float bias[NN];
    __shared__ float tile[NN * NH];        // 16 KB; later overwritten with exp
    __shared__ float pmax[NN], psum[NN], psq[NN];
    __shared__ float mxs[NH], fac[NH];
    __shared__ int   anyFlag;
    if (t == 0) anyFlag = 0;
    __syncthreads();
    {
        const int a = adj[(size_t)bq * NN + t];
        bias[t] = a ? 0.0f : NEG_BIG;
        if (a) atomicOr(&anyFlag, 1);
    }
    __syncthreads();
    if (anyFlag == 0) bias[t] = 0.0f;
    __syncthreads();

    // ---- stage message tile [256 k][16 h] (16 KB) ----
    const char* gbase = (const char*)(imsg + (size_t)bq * NN * NH);
#if HAVE_ASYNC
    for (int i = 0; i < 4; ++i) {
        const int p = i * NN + t;          // 16-byte piece index
        __builtin_amdgcn_global_load_async_to_lds_b128(
            (ASG v4i*)(gbase + (size_t)p * 16),
            (ASL v4i*)(tile + p * 4), 0, 0);
    }
    __builtin_amdgcn_s_wait_asynccnt(0);
#else
    for (int i = 0; i < 4; ++i) {
        const int p = i * NN + t;
        ((float4*)tile)[p] = ((const float4*)gbase)[p];
    }
#endif
    __syncthreads();

    // ---- strip-parallel softmax stats: hs = t&15, strip = t>>4 ----
    const int hs = t & 15, strip = t >> 4;
    float pm = NEG_BIG;
    for (int kk = 0; kk < 16; ++kk) {
        const int k = strip * 16 + kk;
        pm = fmaxf(pm, tile[k * NH + hs] + bias[k]);
    }
    pmax[t] = pm;                          // pmax[strip*16 + hs]
    __syncthreads();
    if (t < NH) {
        float m2 = NEG_BIG;
        for (int j = 0; j < 16; ++j) m2 = fmaxf(m2, pmax[j * 16 + t]);
        mxs[t] = m2;
    }
    __syncthreads();
    const float mh = mxs[hs];
    float ls = 0.0f, lsq = 0.0f;
    for (int kk = 0; kk < 16; ++kk) {
        const int k  = strip * 16 + kk;
        const float em = __expf(tile[k * NH + hs] + bias[k] - mh);
        tile[k * NH + hs] = em;            // overwrite in place
        ls  += em;
        lsq += em * em;
    }
    psum[t] = ls; psq[t] = lsq;
    __syncthreads();
    if (t < NH) {
        float s = 0.0f, sq = 0.0f;
        for (int j = 0; j < 16; ++j) { s += psum[j * 16 + t]; sq += psq[j * 16 + t]; }
        fac[t] = sqrtf(sq) / (s * s);
    }
    __syncthreads();

    // ---- phase B: weighted accumulation, thread = (h = t>>4, e = t&15) ----
    const int h = t >> 4, e = t & 15;
    const float* __restrict__ pp =
        projI + (size_t)(bq >> 8) * NN * DI + h * DH + e;
    float acc = 0.0f;
    for (int k = 0; k < NN; ++k) {
        __builtin_prefetch(pp + (size_t)(k + 16) * DI, 0, 0);
        acc += tile[k * NH + h] * pp[(size_t)k * DI];
    }
    out_pre[(size_t)bq * DI + t] = acc * fac[h];
}

// ---------------------------------------------------------------------------
extern "C" void kernel_launch(void* const* d_in, const int* in_sizes, int n_in,
                              void* d_out, int out_size, void* d_ws, size_t ws_size,
                              hipStream_t stream) {
    (void)in_sizes; (void)n_in; (void)out_size; (void)ws_size;

    const float* v_equi  = (const float*)d_in[0];   // [B,N,3,CE]
    const float* v_inv   = (const float*)d_in[1];   // [B,N,DI]
    const float* emsg    = (const float*)d_in[2];   // [B,N,N,CE]
    const float* imsg    = (const float*)d_in[3];   // [B,N,N,NH]
    const int*   adj     = (const int*)  d_in[4];   // [B,N,N]
    const float* W_coord = (const float*)d_in[5];   // [CE,CE]
    const float* W_eqo   = (const float*)d_in[6];   // [CE,CE]
    const float* b_eqo   = (const float*)d_in[7];   // [CE]
    const float* W_in    = (const float*)d_in[8];   // [DI,DI]
    const float* b_in    = (const float*)d_in[9];   // [DI]
    const float* W_out   = (const float*)d_in[10];  // [DI,DI]
    const float* b_out   = (const float*)d_in[11];  // [DI]

    float* out_equi = (float*)d_out;                       // [B,N,3,CE]
    float* out_inv  = out_equi + (size_t)BB * NN * 3 * CE; // [B,N,DI]

    float* ws    = (float*)d_ws;
    float* projE = ws;                                    // 393216 f32
    float* preE  = projE + (size_t)BB * NN * 3 * CE;      // 393216
    float* projI = preE  + (size_t)BB * NN * 3 * CE;      // 524288
    float* preI  = projI + (size_t)BB * NN * DI;          // 524288

    // --- Equi branch ---
    {
        const int M = BB * NN * 3, K = CE, Nout = CE;          // 6144x64x64
        const int waves = (M / 16) * (Nout / 16);              // 1536
        gemm_wmma_f32<<<waves / 8, 256, 0, stream>>>(v_equi, W_coord, nullptr,
                                                     projE, M, K, Nout);
    }
    equi_core<<<BB * NN, CE, 0, stream>>>(emsg, projE, adj, preE);
    {
        const int M = BB * NN * 3, K = CE, Nout = CE;
        const int waves = (M / 16) * (Nout / 16);
        gemm_wmma_f32<<<waves / 8, 256, 0, stream>>>(preE, W_eqo, b_eqo,
                                                     out_equi, M, K, Nout);
    }

    // --- Inv branch ---
    {
        const int M = BB * NN, K = DI, Nout = DI;              // 2048x256x256
        const int waves = (M / 16) * (Nout / 16);              // 2048
        gemm_wmma_f32<<<waves / 8, 256, 0, stream>>>(v_inv, W_in, b_in,
                                                     projI, M, K, Nout);
    }
    inv_core<<<BB * NN, NN, 0, stream>>>(imsg, projI, adj, preI);
    {
        const int M = BB * NN, K = DI, Nout = DI;
        const int waves = (M / 16) * (Nout / 16);
        gemm_wmma_f32<<<waves / 8, 256, 0, stream>>>(preI, W_out, b_out,
                                                     out_inv, M, K, Nout);
    }
}